// PlaceCellNetwork_7430293422815
// MI455X (gfx1250) — compile-verified
//
#include <hip/hip_runtime.h>

// PlaceCellNetwork fixed-point solve on gfx1250.
// One wave32 owns a 16-row tile. State Z = Y^T (16x16, padded from 10) lives in
// the WMMA C/D layout (8 VGPRs/lane). Per iteration: t = MO^T*Z + (Z - P') via
// three V_WMMA_F32_16X16X4_F32 with the accumulator seeded by (Z - P'), so the
// update is just Z <- relu(Z - dt*t) * rd. B fragments of Z are built with
// immediate-pattern ds_swizzle half-wave swaps. Convergence uses a wave-local
// Frobenius test, divide-free, with a wave-uniform branch (EXEC stays all-ones
// around every WMMA).

typedef float v2f __attribute__((ext_vector_type(2)));
typedef float v8f __attribute__((ext_vector_type(8)));

#define PCN_DT    0.05f
#define PCN_LBD1  0.005f
#define PCN_LBD2  0.005f
#define PCN_TOL   1e-4f
#define PCN_ITERS 50
#define IN_DIM    5
#define OUT_DIM   10

// ds_swizzle group-of-32 pattern: offset = xor<<10 | or<<5 | and
#define SWZ_XOR(m) (((m) << 10) | 0x1f)

template <int MASK>
__device__ __forceinline__ float swz_xor(float x) {
    return __int_as_float(
        __builtin_amdgcn_ds_swizzle(__float_as_int(x), SWZ_XOR(MASK)));
}

__global__ void pcn_kernel(const float* __restrict__ X,
                           const float* __restrict__ W,
                           const float* __restrict__ M,
                           const float* __restrict__ b,
                           float* __restrict__ out,
                           int B)
{
    const int tid    = threadIdx.x;
    const int lane   = tid & 31;
    const int h      = (lane >> 4) & 1;   // wave half
    const int mrow   = lane & 15;
    const int waveId = blockIdx.x * (blockDim.x >> 5) + (tid >> 5);
    const int row0   = waveId * 16;
    if (row0 >= B) return;                // whole-wave uniform exit

    // ---- loop-invariant A fragments: MO^T (zeroed diagonal), padded 16x12 ----
    // A[m=j, k] at (lane = m + 16*(k>>1), vreg = k&1); per chunk c: k = 4c+2h+a
    float amo[3][2];
#pragma unroll
    for (int c = 0; c < 3; ++c)
#pragma unroll
        for (int a = 0; a < 2; ++a) {
            int j = mrow, k = 4 * c + 2 * h + a;
            amo[c][a] = (j < OUT_DIM && k < OUT_DIM && k != j) ? M[k * OUT_DIM + j] : 0.f;
        }

    // ---- A fragments of W (for Wx^T = W * X^T), K = 5 padded to 8 ----
    float aw[2][2];
#pragma unroll
    for (int c = 0; c < 2; ++c)
#pragma unroll
        for (int a = 0; a < 2; ++a) {
            int j = mrow, k = 4 * c + 2 * h + a;
            aw[c][a] = (j < OUT_DIM && k < IN_DIM) ? W[j * IN_DIM + k] : 0.f;
        }

    // ---- B fragments of X^T: B[k,row] = X[row0+row, k] ----
    const int row  = row0 + mrow;
    const int rowc = (row < B) ? row : (B - 1);
    float bx[2][2];
#pragma unroll
    for (int c = 0; c < 2; ++c)
#pragma unroll
        for (int a = 0; a < 2; ++a) {
            int k = 4 * c + 2 * h + a;
            bx[c][a] = (k < IN_DIM) ? X[rowc * IN_DIM + k] : 0.f;
        }

    // ---- per-register constants: j = r + 8h ----
    float rd[8], pb[8];
#pragma unroll
    for (int r = 0; r < 8; ++r) {
        int j = r + 8 * h;
        // v_rcp_f32 (~1 ulp) is plenty for a 1e-4-tolerance fixed point
        rd[r] = (j < OUT_DIM) ? __builtin_amdgcn_rcpf(PCN_LBD2 + M[j * OUT_DIM + j]) : 1.f;
        pb[r] = (j < OUT_DIM) ? b[j] : 0.f;   // sqrt(alpha) == 1
    }

    // ---- Wx^T in C/D layout via two WMMAs ----
    v8f zero = {};
    v2f wa0 = {aw[0][0], aw[0][1]}, wa1 = {aw[1][0], aw[1][1]};
    v2f xb0 = {bx[0][0], bx[0][1]}, xb1 = {bx[1][0], bx[1][1]};
    v8f wx = __builtin_amdgcn_wmma_f32_16x16x4_f32(false, wa1, false, xb1,
                                                   (short)0, zero, false, false);
    wx = __builtin_amdgcn_wmma_f32_16x16x4_f32(false, wa0, false, xb0,
                                               (short)0, wx, false, false);

    // P' = Wx^T - sqrt(a)*b - lbd1/dt  (so relu arg is exactly z - dt*t)
    float p[8], z[8];
#pragma unroll
    for (int r = 0; r < 8; ++r) {
        p[r] = (wx[r] - pb[r]) - (PCN_LBD1 / PCN_DT);
        z[r] = 0.f;
    }

    const v2f am0 = {amo[0][0], amo[0][1]};
    const v2f am1 = {amo[1][0], amo[1][1]};
    const v2f am2 = {amo[2][0], amo[2][1]};
    const bool up = (h != 0);

    // ---- fixed-point loop ----
    for (int it = 0; it < PCN_ITERS; ++it) {
        // B fragments of Z (C/D layout -> B layout needs only half-wave swaps)
        v2f bz0, bz1, bz2;
        float s0 = swz_xor<16>(z[0]), s1 = swz_xor<16>(z[1]);
        float s2 = swz_xor<16>(z[2]), s3 = swz_xor<16>(z[3]);
        float s6 = swz_xor<16>(z[6]), s7 = swz_xor<16>(z[7]);
        bz0.x = up ? s2 : z[0];  bz0.y = up ? s3 : z[1];   // k-block j=0..3
        bz1.x = up ? s6 : z[4];  bz1.y = up ? s7 : z[5];   // k-block j=4..7
        bz2.x = up ? z[2] : s0;  bz2.y = up ? z[3] : s1;   // k-block j=8..11

        // seed accumulator with (Z - P'): t = MO^T Z + Z - P'
        v8f c0;
#pragma unroll
        for (int r = 0; r < 8; ++r) c0[r] = z[r] - p[r];

        v8f t = __builtin_amdgcn_wmma_f32_16x16x4_f32(false, am2, false, bz2,
                                                      (short)0, c0, false, false);
        t = __builtin_amdgcn_wmma_f32_16x16x4_f32(false, am1, false, bz1,
                                                  (short)0, t, false, false);
        t = __builtin_amdgcn_wmma_f32_16x16x4_f32(false, am0, false, bz0,
                                                  (short)0, t, false, false);

        float dacc = 0.f, nacc = 0.f;
#pragma unroll
        for (int r = 0; r < 8; ++r) {
            float zn = fmaxf(fmaf(-PCN_DT, t[r], z[r]), 0.f) * rd[r];
            float d  = zn - z[r];
            dacc = fmaf(d, d, dacc);
            nacc = fmaf(z[r], z[r], nacc);
            z[r] = zn;
        }
        // wave-wide Frobenius reduction via immediate-pattern swizzles
        dacc += swz_xor<1>(dacc);   nacc += swz_xor<1>(nacc);
        dacc += swz_xor<2>(dacc);   nacc += swz_xor<2>(nacc);
        dacc += swz_xor<4>(dacc);   nacc += swz_xor<4>(nacc);
        dacc += swz_xor<8>(dacc);   nacc += swz_xor<8>(nacc);
        dacc += swz_xor<16>(dacc);  nacc += swz_xor<16>(nacc);

        // err < TOL  <=>  dacc < (TOL * (sqrt(nacc) + 1e-10))^2   (divide-free)
        float thr = PCN_TOL * (sqrtf(nacc) + 1e-10f);
        if (dacc < thr * thr) break;      // identical in all lanes -> uniform
    }

    // ---- store Y[row, j] = Z[j, row]; all offsets 8B-aligned -> b64 stores ----
    if (row < B) {
        float* o = out + row * OUT_DIM;
        if (h == 0) {
#pragma unroll
            for (int r = 0; r < 8; r += 2) {         // j = 0..7
                float2 s; s.x = z[r]; s.y = z[r + 1];
                *reinterpret_cast<float2*>(o + r) = s;
            }
        } else {                                     // j = 8,9
            float2 s; s.x = z[0]; s.y = z[1];
            *reinterpret_cast<float2*>(o + 8) = s;
        }
    }
}

extern "C" void kernel_launch(void* const* d_in, const int* in_sizes, int n_in,
                              void* d_out, int out_size, void* d_ws, size_t ws_size,
                              hipStream_t stream) {
    const float* X = (const float*)d_in[0];
    const float* W = (const float*)d_in[1];
    const float* M = (const float*)d_in[2];
    const float* b = (const float*)d_in[3];
    float* out = (float*)d_out;

    const int B     = in_sizes[0] / IN_DIM;       // 500000
    const int tiles = (B + 15) / 16;              // one wave per 16 rows
    const int wavesPerBlock = 8;                  // 256 threads = 8 wave32
    const int blocks = (tiles + wavesPerBlock - 1) / wavesPerBlock;

    pcn_kernel<<<blocks, wavesPerBlock * 32, 0, stream>>>(X, W, M, b, out, B);
}